// GG_84705345012339
// MI455X (gfx1250) — compile-verified
//
#include <hip/hip_runtime.h>

// Problem constants (from the reference)
#define N_NODES 30000
#define S_NBR   100
#define DIN     512
#define DOUT    128

// M-blocking: 3 x 16-row WMMA tiles per block. 30000 / 48 = 625, exact.
// Each B fragment load now feeds 3 WMMAs -> B L2 traffic cut 3x (480->160MB).
#define MTILES  3
#define MBLOCK  (MTILES * 16)

typedef float v2f __attribute__((ext_vector_type(2)));
typedef float v4f __attribute__((ext_vector_type(4)));
typedef float v8f __attribute__((ext_vector_type(8)));

// LDS row stride for the A tile: 512 would put all 16 column-read lanes on
// the same bank (512 % 64 == 0).  516 % 64 == 4 -> banks kk + 4m, distinct
// for m = 0..15 -> conflict-free.  48 * 516 * 4B = ~97 KB (3 blocks/WGP).
#define LDS_STRIDE 516

// ---------------------------------------------------------------------------
// Kernel 1: support = x @ W via V_WMMA_F32_16X16X4_F32 (fp32 in/out, matches
// reference precision).  One block = 48-row M block; 8 waves cover the 128
// output columns (16 cols per wave); each wave runs 3 independent WMMA
// accumulator chains over the shared B fragment.
// ---------------------------------------------------------------------------
__global__ __launch_bounds__(256) void gemm_wmma(const float* __restrict__ x,
                                                 const float* __restrict__ W,
                                                 float* __restrict__ support) {
  __shared__ float lds_a[MBLOCK * LDS_STRIDE];

  const int m0 = blockIdx.x * MBLOCK;      // 625 blocks, exact
  const int t  = threadIdx.x;

  // Cooperative load of the 48x512 A tile: 24576 floats, 256 threads,
  // 24 x float4 each.  Global side fully coalesced; LDS side 16B aligned
  // (516 % 4 == 0 and col % 4 == 0).
#pragma unroll
  for (int i = 0; i < (MBLOCK * DIN) / (256 * 4); ++i) {
    const int e   = (i * 256 + t) * 4;     // flat element in 48x512 tile
    const int row = e >> 9;                // / 512
    const int col = e & 511;
    v4f v = *(const v4f*)(x + (size_t)(m0 + row) * DIN + col);
    *(v4f*)(lds_a + row * LDS_STRIDE + col) = v;
  }
  __syncthreads();

  const int wave = t >> 5;
  const int lane = t & 31;
  const int m    = lane & 15;              // row within tile (A), col (B/C/D)
  const int g    = lane >> 4;              // lane group: 0 or 1
  const int n0   = wave * 16;              // this wave's N tile

  v8f acc0 = {0.f, 0.f, 0.f, 0.f, 0.f, 0.f, 0.f, 0.f};
  v8f acc1 = acc0;
  v8f acc2 = acc0;

  const float* a0p  = lds_a + m * LDS_STRIDE;          // tile 0 row
  const float* a1p  = a0p + 16 * LDS_STRIDE;           // tile 1 row
  const float* a2p  = a0p + 32 * LDS_STRIDE;           // tile 2 row
  const float* wcol = W + n0 + m;                      // B column for this lane

#pragma unroll 4
  for (int k = 0; k < DIN; k += 4) {
    const int ka = k + 2 * g;              // ISA layout: VGPR i, group g -> K = 2g+i
    v2f b;                                 // B: row K striped across lanes
    b.x = wcol[(size_t)ka * DOUT];
    b.y = wcol[(size_t)(ka + 1) * DOUT];

    v2f a;
    a.x = a0p[ka]; a.y = a0p[ka + 1];
    acc0 = __builtin_amdgcn_wmma_f32_16x16x4_f32(false, a, false, b,
                                                 (short)0, acc0, false, false);
    a.x = a1p[ka]; a.y = a1p[ka + 1];
    acc1 = __builtin_amdgcn_wmma_f32_16x16x4_f32(false, a, false, b,
                                                 (short)0, acc1, false, false);
    a.x = a2p[ka]; a.y = a2p[ka + 1];
    acc2 = __builtin_amdgcn_wmma_f32_16x16x4_f32(false, a, false, b,
                                                 (short)0, acc2, false, false);
  }

  // C/D layout: lanes 0-15 VGPR v -> M=v ; lanes 16-31 VGPR v -> M=v+8
  float* outp = support + (size_t)m0 * DOUT + n0 + m;
#pragma unroll
  for (int v = 0; v < 8; ++v) {
    const size_t r = (size_t)(v + 8 * g);
    outp[r * DOUT]                = acc0[v];
    outp[(r + 16) * DOUT]         = acc1[v];
    outp[(r + 32) * DOUT]         = acc2[v];
  }
}

// ---------------------------------------------------------------------------
// Kernel 2: out[n] = (support[n] + sum_s support[nbr[n][s]]) / 101 + bias
// One wave32 per node; lane l owns float4 columns [4l, 4l+4).  The gather is
// L2-resident (support table = 15.4 MB << 192 MB L2); 8 row-gathers in
// flight (unroll 2 x 4), two accumulator chains to shorten the FP dep path.
// ---------------------------------------------------------------------------
__global__ __launch_bounds__(256) void sage_mean(const float* __restrict__ support,
                                                 const int* __restrict__ nbr,
                                                 const float* __restrict__ bias,
                                                 float* __restrict__ out) {
  const int lane = threadIdx.x & 31;
  const int node = blockIdx.x * 8 + (threadIdx.x >> 5);   // 3750 blocks, exact

  const v4f* table = (const v4f*)support;                 // row r = table[r*32 + lane]
  v4f acc0 = table[(size_t)node * 32 + lane];             // self row
  v4f acc1 = {0.f, 0.f, 0.f, 0.f};

  const int* ni = nbr + (size_t)node * S_NBR;
#pragma unroll 2
  for (int s = 0; s < S_NBR; s += 4) {
    const int i0 = __builtin_amdgcn_readfirstlane(ni[s + 0]);
    const int i1 = __builtin_amdgcn_readfirstlane(ni[s + 1]);
    const int i2 = __builtin_amdgcn_readfirstlane(ni[s + 2]);
    const int i3 = __builtin_amdgcn_readfirstlane(ni[s + 3]);
    v4f r0 = table[(size_t)i0 * 32 + lane];
    v4f r1 = table[(size_t)i1 * 32 + lane];
    v4f r2 = table[(size_t)i2 * 32 + lane];
    v4f r3 = table[(size_t)i3 * 32 + lane];
    acc0 += (r0 + r1);
    acc1 += (r2 + r3);
  }

  const v4f bb  = *(const v4f*)(bias + lane * 4);
  const v4f res = (acc0 + acc1) * (1.0f / (float)(S_NBR + 1)) + bb;
  *(v4f*)(out + (size_t)node * DOUT + lane * 4) = res;
}

// ---------------------------------------------------------------------------
extern "C" void kernel_launch(void* const* d_in, const int* in_sizes, int n_in,
                              void* d_out, int out_size, void* d_ws, size_t ws_size,
                              hipStream_t stream) {
  const float* x   = (const float*)d_in[0];   // [30000, 512]
  const float* W   = (const float*)d_in[1];   // [512, 128]
  const float* b   = (const float*)d_in[2];   // [128]
  const int*   nbr = (const int*)d_in[3];     // [30000, 100]
  float* out     = (float*)d_out;             // [30000, 128]
  float* support = (float*)d_ws;              // scratch: 30000*128*4 = 15.36 MB

  gemm_wmma<<<N_NODES / MBLOCK, 256, 0, stream>>>(x, W, support);
  sage_mean<<<N_NODES / 8, 256, 0, stream>>>(support, nbr, b, out);
}